// ConvCaps_19705309954634
// MI455X (gfx1250) — compile-verified
//
#include <hip/hip_runtime.h>

typedef float v2f __attribute__((ext_vector_type(2)));
typedef float v8f __attribute__((ext_vector_type(8)));
typedef unsigned int v4u __attribute__((ext_vector_type(4)));
typedef int v4i __attribute__((ext_vector_type(4)));
typedef int v8i __attribute__((ext_vector_type(8)));

#define NPOS   508
#define NB     288
#define NC     32
#define PS     16
#define EPSF   1e-8f
#define LAM    1e-3f
#define LN2PI  1.8378770664093453f

// LDS layout (float offsets):
//   XROW : 9*544 = 4896 floats (raw input rows incl. activations, TDM target)
//   MU/SIG/INV2 : 512 each ; RSUM/CB/AOUT : 32 each
//   VH   : 147456 halves (v as f16)  -> total 321024 bytes (<= 320KB WGP LDS)
#define OFF_MU   4896
#define OFF_SIG  5408
#define OFF_INV2 5920
#define OFF_RSUM 6432
#define OFF_CB   6464
#define OFF_AOUT 6496
#define OFF_VH   6528
#define SMEM_BYTES (OFF_VH * 4 + 147456 * 2)   // 321024

__device__ __forceinline__ float warp_sum(float v) {
#pragma unroll
    for (int m = 16; m > 0; m >>= 1) v += __shfl_xor(v, m, 32);
    return v;
}
__device__ __forceinline__ float warp_max(float v) {
#pragma unroll
    for (int m = 16; m > 0; m >>= 1) v = fmaxf(v, __shfl_xor(v, m, 32));
    return v;
}

// Issue one TDM load: 544 contiguous f32 from global -> LDS byte offset lds_off.
__device__ __forceinline__ void tdm_load_row(const float* src, unsigned lds_off) {
    unsigned long long ga = (unsigned long long)(uintptr_t)src;
    v4u g0;
    g0[0] = 1u;                                   // count=1, user mode, no gather
    g0[1] = lds_off;                              // lds_addr
    g0[2] = (unsigned)(ga & 0xffffffffu);         // global_addr[31:0]
    g0[3] = (unsigned)((ga >> 32) & 0x01ffffffu)  // global_addr[56:32]
            | 0x80000000u;                        // type=2 ("image") in [127:126]
    v8i g1;
    g1[0] = 0x20000;                              // data_size=2 (4B), mask=0
    g1[1] = (int)(544u << 16);                    // tensor_dim0[15:0] in [63:48]
    g1[2] = (int)(1u << 16);                      // tensor_dim1[15:0] = 1
    g1[3] = (int)(544u << 16);                    // tile_dim0 = 544 in [127:112]
    g1[4] = 1;                                    // tile_dim1 = 1
    g1[5] = 544;                                  // tensor_dim0_stride = 544
    g1[6] = 0;
    g1[7] = 0;
    v4i z4 = {0, 0, 0, 0};
#if defined(__clang_major__) && (__clang_major__ >= 23)
    v8i z8 = {0, 0, 0, 0, 0, 0, 0, 0};
    __builtin_amdgcn_tensor_load_to_lds(g0, g1, z4, z4, z8, 0);
#else
    __builtin_amdgcn_tensor_load_to_lds(g0, g1, z4, z4, 0);
#endif
}

__global__ void __launch_bounds__(512, 1)
convcaps_em_kernel(const float* __restrict__ x,
                   const float* __restrict__ Wp,      // (288,32,4,4)
                   const float* __restrict__ beta_u,  // (32)
                   const float* __restrict__ beta_a,  // (32)
                   float* __restrict__ out,           // (508*544)
                   float* __restrict__ r_out)         // (508*288*32)
{
    extern __shared__ float smem[];
    float*    XROW = smem;
    float*    MU   = smem + OFF_MU;
    float*    SIG  = smem + OFF_SIG;
    float*    INV2 = smem + OFF_INV2;
    float*    RSUM = smem + OFF_RSUM;
    float*    CB   = smem + OFF_CB;
    float*    AOUT = smem + OFF_AOUT;
    _Float16* VH   = (_Float16*)(smem + OFF_VH);

    const int n    = blockIdx.x;       // routing problem id (n = b*254 + ow for output)
    const int t    = threadIdx.x;      // 0..511
    const int w    = t >> 5;           // wave id 0..15
    const int lane = t & 31;           // c index for routing

    // ------- Stage A1: TDM the 9 input rows this n needs into LDS ----------
    if (w == 0) {
#pragma unroll
        for (int q = 0; q < 9; ++q) {
            int CI = n * 9 + q;                         // scrambled cell id
            int b  = CI / 2286;  int r1 = CI - b * 2286;
            int kx = r1 / 762;   int r2 = r1 - kx * 762;
            int ow = r2 % 254;
            const float* src = x + (size_t)((b * 256 + ow + kx) * 4) * 544;  // h = 0
            tdm_load_row(src, (unsigned)(q * 544 * 4));
        }
        __builtin_amdgcn_s_wait_tensorcnt(0);
    }
    __syncthreads();

    // ------- Stage A2: pose transform via V_WMMA_F32_16X16X4_F32 -----------
    // Wave handles B = w (mod 16): 18 capsules x 8 column tiles of (c,k).
    const int rr = lane & 3;             // A row (rows 4..15 are duplicates -> D rows discarded)
    const int cl = lane & 15;            // B/D column within tile
    const int j0 = (lane >> 4) << 1;     // K offset: lanes 0-15 -> {0,1}, 16-31 -> {2,3}
    for (int B = w; B < NB; B += 16) {
        const int q  = B >> 5;
        const int Bl = B & 31;
        v2f a;
        a.x = XROW[q * 544 + Bl * 16 + rr * 4 + j0];
        a.y = XROW[q * 544 + Bl * 16 + rr * 4 + j0 + 1];
        if (B + 16 < NB) __builtin_prefetch(Wp + (B + 16) * 512, 0, 0);
#pragma unroll
        for (int ct = 0; ct < 8; ++ct) {
            const int gcol = ct * 16 + cl;
            const int c = gcol >> 2, k = gcol & 3;
            const int wbase = B * 512 + c * 16 + k;
            v2f bm;
            bm.x = Wp[wbase + j0 * 4];
            bm.y = Wp[wbase + (j0 + 1) * 4];
            v8f acc = {};
            v8f d = __builtin_amdgcn_wmma_f32_16x16x4_f32(
                false, a, false, bm, (short)0, acc, false, false);
            // D rows 0..3 live in d[0..3], lanes 0..15; store v[B][c][i*4+k] as f16
            if (lane < 16) {
                const int vb = B * 512 + c * 16 + k;
                VH[vb + 0]  = (_Float16)d[0];
                VH[vb + 4]  = (_Float16)d[1];
                VH[vb + 8]  = (_Float16)d[2];
                VH[vb + 12] = (_Float16)d[3];
            }
        }
    }
    __syncthreads();

    // ------- Stage B: EM routing -------------------------------------------
    // thread (w, lane) owns r[B][c] for B in {16j + w}, c = lane.
    float r[18], rh[18], ain[18];
#pragma unroll
    for (int j = 0; j < 18; ++j) {
        const int Bj = j * 16 + w;
        r[j]   = 1.0f / 32.0f;
        ain[j] = XROW[(Bj >> 5) * 544 + 512 + (Bj & 31)];
    }

    for (int it = 0; it < 3; ++it) {
        __syncthreads();
        MU[t]  = 0.0f;
        SIG[t] = 0.0f;
        if (t < 32) RSUM[t] = 0.0f;
        __syncthreads();

        // m_step: r_hat = (r*a_in) normalized over c; rsum[c] = sum_B r_hat
        float part = 0.0f;
#pragma unroll
        for (int j = 0; j < 18; ++j) {
            float v = r[j] * ain[j];
            float s = warp_sum(v);
            rh[j] = v / (s + EPSF);
            part += rh[j];
        }
        atomicAdd(&RSUM[lane], part);
        __syncthreads();

        const float rs_l = RSUM[lane] + EPSF;

        // mu accumulation
#pragma unroll
        for (int j = 0; j < 18; ++j) {
            const float coeff = rh[j] / rs_l;
            const int vb = (j * 16 + w) * 512 + lane * 16;
#pragma unroll
            for (int p = 0; p < 16; ++p)
                atomicAdd(&MU[lane * 16 + p], coeff * (float)VH[vb + p]);
        }
        __syncthreads();

        // sigma accumulation
#pragma unroll
        for (int j = 0; j < 18; ++j) {
            const float coeff = rh[j] / rs_l;
            const int vb = (j * 16 + w) * 512 + lane * 16;
#pragma unroll
            for (int p = 0; p < 16; ++p) {
                float dv = (float)VH[vb + p] - MU[lane * 16 + p];
                atomicAdd(&SIG[lane * 16 + p], coeff * dv * dv);
            }
        }
        __syncthreads();
        { float s = SIG[t] + EPSF; SIG[t] = s; INV2[t] = 0.5f / s; }
        __syncthreads();

        // a_out / per-c bias terms
        if (t < 32) {
            const float rs = RSUM[t];
            float clog = 0.0f;
#pragma unroll
            for (int p = 0; p < 16; ++p) clog += 0.5f * __logf(SIG[t * 16 + p]);
            const float cost = rs * (16.0f * beta_u[t] + clog);
            const float ao = 1.0f / (1.0f + __expf(-LAM * (beta_a[t] - cost)));
            AOUT[t] = ao;
            CB[t]   = __logf(ao) - clog - 8.0f * LN2PI;
        }
        __syncthreads();

        // e_step (skipped on last iteration): softmax over c = lanes
        if (it < 2) {
#pragma unroll
            for (int j = 0; j < 18; ++j) {
                const int vb = (j * 16 + w) * 512 + lane * 16;
                float accv = CB[lane];
#pragma unroll
                for (int p = 0; p < 16; ++p) {
                    float dv = (float)VH[vb + p] - MU[lane * 16 + p];
                    accv -= dv * dv * INV2[lane * 16 + p];
                }
                const float mx = warp_max(accv);
                const float e  = __expf(accv - mx);
                const float s  = warp_sum(e);
                r[j] = e / s;
            }
        }
    }

    // ------- outputs --------------------------------------------------------
    out[(size_t)n * 544 + t] = MU[t];                 // mu: channels 0..511
    if (t < 32) out[(size_t)n * 544 + 512 + t] = AOUT[t];
#pragma unroll
    for (int j = 0; j < 18; ++j)
        r_out[((size_t)n * NB + (j * 16 + w)) * 32 + lane] = r[j];
}

extern "C" void kernel_launch(void* const* d_in, const int* in_sizes, int n_in,
                              void* d_out, int out_size, void* d_ws, size_t ws_size,
                              hipStream_t stream) {
    const float* x  = (const float*)d_in[0];
    const float* Wp = (const float*)d_in[1];
    const float* bu = (const float*)d_in[2];
    const float* ba = (const float*)d_in[3];
    float* out   = (float*)d_out;
    float* r_out = out + (size_t)2 * 254 * 544;   // tuple element 2 of the reference

    convcaps_em_kernel<<<NPOS, 512, SMEM_BYTES, stream>>>(x, Wp, bu, ba, out, r_out);
}